// ConstrainedSoftmax_13907104105178
// MI455X (gfx1250) — compile-verified
//
#include <hip/hip_runtime.h>
#include <math.h>

#define KDIM 1024
#define BLK  1024

typedef __attribute__((address_space(3))) float lds_f;

__global__ __launch_bounds__(BLK) void constrained_softmax_kernel(
    const float* __restrict__ z, const float* __restrict__ u,
    float* __restrict__ out) {
  __shared__ float sE[KDIM];    // e values, original order
  __shared__ float sU[KDIM];    // u values, original order
  __shared__ float sKey[KDIM];  // staging for z, then sort keys, then p (original order)
  __shared__ int   sIdx[KDIM];  // permutation payload
  __shared__ float sWaveE[32];  // cross-wave scan of e
  __shared__ float sWaveU[32];  // cross-wave scan of u
  __shared__ float sRed[32];
  __shared__ float sB[2];
  __shared__ int   sM;

  const int  tid  = threadIdx.x;
  const int  lane = tid & 31;
  const int  wave = tid >> 5;
  const long base = (long)blockIdx.x * KDIM;

  if (tid == 0) sM = KDIM;   // init before first barrier (no extra sync later)

  // ---- CDNA5 async global->LDS copy of this row (tracked by ASYNCcnt) ----
  {
    unsigned lz = (unsigned)(size_t)(lds_f*)(&sKey[tid]);  // stage z into sKey
    unsigned lu = (unsigned)(size_t)(lds_f*)(&sU[tid]);
    const float* gz = z + base + tid;
    const float* gu = u + base + tid;
    asm volatile("global_load_async_to_lds_b32 %0, %1, off"
                 :: "v"(lz), "v"(gz) : "memory");
    asm volatile("global_load_async_to_lds_b32 %0, %1, off"
                 :: "v"(lu), "v"(gu) : "memory");
    asm volatile("s_wait_asynccnt 0" ::: "memory");
  }
  __syncthreads();

  const float zt = sKey[tid];
  const float ut = sU[tid];

  // ---- row max (wave32 shuffle tree + one cross-wave LDS hop) ----
  {
    float m = zt;
    for (int o = 16; o; o >>= 1) m = fmaxf(m, __shfl_xor(m, o, 32));
    if (lane == 0) sRed[wave] = m;
    __syncthreads();
    if (tid < 32) {
      float v = sRed[tid];
      for (int o = 16; o; o >>= 1) v = fmaxf(v, __shfl_xor(v, o, 32));
      if (tid == 0) sB[0] = v;
    }
    __syncthreads();
  }
  const float zmax = sB[0];
  const float e = expf(zt - zmax);
  sE[tid] = e;

  // ---- bitonic sort ascending on key = -(e/u)  (== descending by e/u) ----
  float key = -(e / ut);
  int   idx = tid;

  // register/shuffle compare-exchange (intra-wave, stride j <= 16)
  auto cmpx = [&](int j, bool up) {
    float kp = __shfl_xor(key, j, 32);
    int   ip = __shfl_xor(idx, j, 32);
    bool lower   = ((tid & j) == 0);
    bool wantMin = (lower == up);
    bool take    = wantMin ? (kp < key) : (kp > key);
    if (take) { key = kp; idx = ip; }
  };

  // Phase A: k = 2..32 fully in registers, zero barriers
  for (int k = 2; k <= 32; k <<= 1) {
    bool up = ((tid & k) == 0);
    for (int j = k >> 1; j >= 1; j >>= 1) cmpx(j, up);
  }

  // Phase B: k = 64..1024; strides >=32 via LDS, strides <=16 via shuffles
  for (int k = 64; k <= KDIM; k <<= 1) {
    sKey[tid] = key; sIdx[tid] = idx;
    __syncthreads();
    for (int j = k >> 1; j >= 32; j >>= 1) {
      int ixj = tid ^ j;
      if (ixj > tid) {
        float a = sKey[tid], b = sKey[ixj];
        bool asc = ((tid & k) == 0);
        if (asc ? (a > b) : (a < b)) {
          sKey[tid] = b; sKey[ixj] = a;
          int ia = sIdx[tid]; sIdx[tid] = sIdx[ixj]; sIdx[ixj] = ia;
        }
      }
      __syncthreads();
    }
    key = sKey[tid]; idx = sIdx[tid];
    bool up = ((tid & k) == 0);
    for (int j = 16; j >= 1; j >>= 1) cmpx(j, up);
  }

  // ---- gather sorted e,u; wave shuffle scans + one cross-wave scan ----
  const float es = sE[idx];
  const float us = sU[idx];
  float eInc = es, uInc = us;
  for (int o = 1; o < 32; o <<= 1) {
    float te = __shfl_up(eInc, o, 32);
    float tu = __shfl_up(uInc, o, 32);
    if (lane >= o) { eInc += te; uInc += tu; }
  }
  if (lane == 31) { sWaveE[wave] = eInc; sWaveU[wave] = uInc; }
  __syncthreads();
  if (tid < 32) {
    float ve = sWaveE[tid], vu = sWaveU[tid];
    for (int o = 1; o < 32; o <<= 1) {
      float te = __shfl_up(ve, o, 32);
      float tu = __shfl_up(vu, o, 32);
      if (lane >= o) { ve += te; vu += tu; }
    }
    sWaveE[tid] = ve; sWaveU[tid] = vu;   // inclusive wave totals
  }
  __syncthreads();
  const float Z0    = sWaveE[31];                       // grand total of e
  const float baseE = (wave > 0) ? sWaveE[wave - 1] : 0.0f;
  const float baseU = (wave > 0) ? sWaveU[wave - 1] : 0.0f;
  const float Eexcl = (eInc + baseE) - es;
  const float Uexcl = (uInc + baseU) - us;

  // ---- capped-prefix test; m = first uncapped sorted position ----
  float Zi = Z0 - Eexcl;
  if (fabsf(Zi) < 5e-13f) Zi = 1e-6f;
  const float vali   = es * (1.0f - Uexcl) / Zi;
  const bool  capped = vali > us;

  int cand = capped ? KDIM : tid;
  for (int o = 16; o; o >>= 1) cand = min(cand, __shfl_xor(cand, o, 32));
  if (lane == 0) atomicMin(&sM, cand);   // one ds_min per wave
  __syncthreads();
  const int m_ = sM;

  if (tid == m_) {
    float Zm = Z0 - Eexcl;
    if (fabsf(Zm) < 5e-13f) Zm = 1e-6f;
    sB[1] = (1.0f - Uexcl) / Zm;         // tail scale (1 - s_m) / Z_m
  }
  __syncthreads();
  const float scale = (m_ < KDIM) ? sB[1] : 0.0f;

  // ---- produce p, scatter to original order in LDS, store coalesced ----
  const float p = (tid < m_) ? us : es * scale;
  sKey[idx] = p;
  __syncthreads();
  out[base + tid] = sKey[tid];
}

extern "C" void kernel_launch(void* const* d_in, const int* in_sizes, int n_in,
                              void* d_out, int out_size, void* d_ws, size_t ws_size,
                              hipStream_t stream) {
  (void)n_in; (void)out_size; (void)d_ws; (void)ws_size;
  const float* z = (const float*)d_in[0];
  const float* u = (const float*)d_in[1];
  float* out = (float*)d_out;
  int rows = in_sizes[0] / KDIM;
  if (rows <= 0) return;
  constrained_softmax_kernel<<<dim3(rows), dim3(BLK), 0, stream>>>(z, u, out);
}